// LoRAConvsByRandom_cu_clone_78477642433086
// MI455X (gfx1250) — compile-verified
//
#include <hip/hip_runtime.h>
#include <stdint.h>

// ---- problem constants (from the reference) -------------------------------
#define NKc        11          // ceil(51/5)
#define HINc       68          // 64 + PAD(4)
#define CH_WORDS   (HINc*HINc) // 4624 floats per channel plane
#define SLAB_WORDS (NKc*CH_WORDS) // 50864 floats = 203,456 B  (< 320 KB LDS)
#define C_OUTc     64
#define C_INc      (C_OUTc*NKc)   // 704
#define Bc         16
#define OHc        64
#define OWc        64
#define PLANEc     (OHc*OWc)           // 4096
#define OUT_TOTAL  (Bc*C_OUTc*PLANEc)  // 4,194,304 per output tensor
// SHIFTS[i] = 21 - 5i  ->  src coord = out + 5i - 21 ;  EXTRA = 2

typedef unsigned int u32x4 __attribute__((ext_vector_type(4)));
typedef int          i32x4 __attribute__((ext_vector_type(4)));
typedef int          i32x8 __attribute__((ext_vector_type(8)));
typedef float        f32x2 __attribute__((ext_vector_type(2)));   // native vector:
                                                                  // legal for NT stores

// ---------------------------------------------------------------------------
// TDM: DMA one contiguous (11 x 68 x 68) f32 slab from global into LDS.
// Descriptor bitfields per CDNA5 ISA §8.3-8.6 (D# groups 0/1; groups 2/3 zero).
// 2D tile: tile_dim0 = 4624 elems, tile_dim1 = 11, stride = 4624, 4B elems.
// ---------------------------------------------------------------------------
__device__ __forceinline__ void tdm_load_slab(const float* gsrc, uint32_t lds_byte_off) {
  uint64_t ga = (uint64_t)(uintptr_t)gsrc;
  u32x4 g0;
  g0[0] = 1u;                                              // count=1 (valid), user mode
  g0[1] = lds_byte_off;                                    // lds_addr (bytes)
  g0[2] = (uint32_t)(ga & 0xFFFFFFFFu);                    // global_addr[31:0]
  g0[3] = (uint32_t)((ga >> 32) & 0x1FFFFFFu) | (2u << 30);// global_addr[56:32] | type=2
  i32x8 g1;
  g1[0] = (int)(2u << 16);                                 // data_size=2 (4B); mask/pad/iter=0
  g1[1] = (int)((uint32_t)CH_WORDS << 16);                 // tensor_dim0[15:0]=4624
  g1[2] = (int)((uint32_t)NKc      << 16);                 // tensor_dim0 hi=0 | tensor_dim1[15:0]=11
  g1[3] = (int)((uint32_t)CH_WORDS << 16);                 // tensor_dim1 hi=0 | tile_dim0=4624
  g1[4] = NKc;                                             // tile_dim1=11, tile_dim2=0
  g1[5] = CH_WORDS;                                        // tensor_dim0_stride lo32 = 4624
  g1[6] = (int)((uint32_t)CH_WORDS << 16);                 // stride0 hi=0 | tensor_dim1_stride lo16=4624
  g1[7] = 0;                                               // tensor_dim1_stride hi32
  i32x4 gz = {0, 0, 0, 0};
#if defined(__clang_major__) && (__clang_major__ >= 23)
  i32x8 gz8 = {0, 0, 0, 0, 0, 0, 0, 0};
  __builtin_amdgcn_tensor_load_to_lds(g0, g1, gz, gz, gz8, 0);
#else
  __builtin_amdgcn_tensor_load_to_lds(g0, g1, gz, gz, 0);
#endif
}

// ---------------------------------------------------------------------------
// One workgroup per (batch b, group c).  256 threads = 8 wave32.
// lane (t&31) -> column pair w0 = 2*lane ; wave (t>>5) -> row; 8 row passes.
// ---------------------------------------------------------------------------
extern "C" __global__ __launch_bounds__(256, 1)
void lora_shift_kernel(const float* __restrict__ inputs,
                       const int*   __restrict__ idx1,
                       const int*   __restrict__ idx2,
                       const int*   __restrict__ idx_small,
                       float* __restrict__ out1,
                       float* __restrict__ out2,
                       float* __restrict__ outS) {
  extern __shared__ float lds[];       // SLAB_WORDS floats (dynamic, 203,456 B)
  __shared__ int s_o1[4 * NKc];        // per-(r,i) LDS word offsets, lora1 perm
  __shared__ int s_o2[4 * NKc];        // lora2 perm
  __shared__ int s_oS[4];              // small perm

  const int c = blockIdx.x;            // 0..63  group
  const int b = blockIdx.y;            // 0..15  batch
  const int t = threadIdx.x;

  // ---- wave 0: launch the TDM DMA of the whole group slab ----------------
  if (t < 32) {
    const float* src = inputs + ((size_t)b * C_INc + (size_t)c * NKc) * CH_WORDS;
    tdm_load_slab(src, (uint32_t)(uintptr_t)lds);
  }

  // ---- meanwhile: fetch the 92 permutation indices for this group --------
  if (t < 4 * NKc) {
    int r = t / NKc, i = t - r * NKc;
    s_o1[t] = (idx1[r * C_INc + c * NKc + i] - c * NKc) * CH_WORDS;
  } else if (t < 8 * NKc) {
    int j = t - 4 * NKc;
    int r = j / NKc, i = j - r * NKc;
    s_o2[j] = (idx2[r * C_INc + c * NKc + i] - c * NKc) * CH_WORDS;
  } else if (t < 8 * NKc + 4) {
    int r = t - 8 * NKc;
    s_oS[r] = (idx_small[r * C_OUTc + c] - c * NKc) * CH_WORDS;
  }

  if (t < 32) __builtin_amdgcn_s_wait_tensorcnt(0);   // wave 0 drains its TDM op
  __syncthreads();                                    // publish LDS to all waves

  // registerize channel offsets (1 block/WGP due to LDS -> VGPRs plentiful)
  int o1[4 * NKc], o2[4 * NKc], oS[4];
#pragma unroll
  for (int j = 0; j < 4 * NKc; ++j) { o1[j] = s_o1[j]; o2[j] = s_o2[j]; }
#pragma unroll
  for (int r = 0; r < 4; ++r) oS[r] = s_oS[r];

  const int    w0    = (t & 31) * 2;   // even column -> 8B-aligned LDS pairs
  const int    hv    = t >> 5;         // wave-uniform row id
  const size_t obase = ((size_t)b * C_OUTc + c) * PLANEc;

  for (int hh = 0; hh < 8; ++hh) {
    const int h    = hv + 8 * hh;
    const int row2 = (h + 2) * HINc;                 // lora2 / small source row
    f32x2 a1 = {0.f, 0.f};
    f32x2 a2 = {0.f, 0.f};

#pragma unroll
    for (int i = 0; i < NKc; ++i) {
      // ---- lora1: row y = h + 5i - 21, cols (w0+2, w0+3), aligned b64 ----
      const int  y  = h + 5 * i - 21;
      const bool vy = (unsigned)y < (unsigned)HINc;          // wave-uniform
      const int  r1 = (vy ? y : 0) * HINc + (w0 + 2);
      // ---- lora2: row h+2, cols xx = w0 + 5i - 21 and xx+1 ---------------
      const int  xx = w0 + 5 * i - 21;
      if (i & 1) {
        // i odd -> xx even -> pair is 8B-aligned and jointly valid/invalid:
        //   xx<0  => xx<=-2 => xx+1<0 ;  xx>66 => xx>=68 => both OOB.
        const bool vp = (unsigned)xx < (unsigned)HINc;       // == (0<=xx<=66)
        const int  xc = vp ? xx : 0;
#pragma unroll
        for (int r = 0; r < 4; ++r) {
          const f32x2 v1 = *(const f32x2*)(lds + o1[r * NKc + i] + r1);
          if (vy) a1 += v1;                                  // v_pk_add_f32 candidate
          const f32x2 v2 = *(const f32x2*)(lds + o2[r * NKc + i] + row2 + xc);
          if (vp) a2 += v2;
        }
      } else {
        // i even -> xx odd -> misaligned pair: two predicated scalar loads
        const bool vx0 = (unsigned)xx       < (unsigned)HINc;
        const bool vx1 = (unsigned)(xx + 1) < (unsigned)HINc;
        const int  x0  = vx0 ? xx       : 0;
        const int  x1  = vx1 ? (xx + 1) : 0;
#pragma unroll
        for (int r = 0; r < 4; ++r) {
          const f32x2 v1 = *(const f32x2*)(lds + o1[r * NKc + i] + r1);
          if (vy) a1 += v1;
          const float v20 = lds[o2[r * NKc + i] + row2 + x0];
          const float v21 = lds[o2[r * NKc + i] + row2 + x1];
          if (vx0) a2.x += v20;
          if (vx1) a2.y += v21;
        }
      }
    }

    f32x2 as = {0.f, 0.f};
#pragma unroll
    for (int r = 0; r < 4; ++r) {
      const f32x2 v = *(const f32x2*)(lds + oS[r] + row2 + (w0 + 2));
      as += v;
    }

    const size_t o = obase + (size_t)h * OWc + w0;
    __builtin_nontemporal_store(a1, (f32x2*)(out1 + o));   // global_store_b64 NT
    __builtin_nontemporal_store(a2, (f32x2*)(out2 + o));
    __builtin_nontemporal_store(as, (f32x2*)(outS + o));
  }
}

// ---------------------------------------------------------------------------
extern "C" void kernel_launch(void* const* d_in, const int* in_sizes, int n_in,
                              void* d_out, int out_size, void* d_ws, size_t ws_size,
                              hipStream_t stream) {
  const float* inputs   = (const float*)d_in[0];   // (16, 704, 68, 68) f32
  const int*   idx1     = (const int*)d_in[1];     // (4, 704) i32
  const int*   idx2     = (const int*)d_in[2];     // (4, 704) i32
  const int*   idxS     = (const int*)d_in[3];     // (4, 64)  i32
  float*       out      = (float*)d_out;           // lora1 | lora2 | small

  dim3 grid(C_OUTc, Bc);       // one workgroup per (group, batch)
  dim3 block(256);
  size_t shmem = (size_t)SLAB_WORDS * sizeof(float);   // 203,456 B of LDS

  hipLaunchKernelGGL(lora_shift_kernel, grid, block, shmem, stream,
                     inputs, idx1, idx2, idxS,
                     out, out + OUT_TOTAL, out + 2 * (size_t)OUT_TOTAL);
}